// BEVDistillationMaskGenerator_label_88115549045404
// MI455X (gfx1250) — compile-verified
//
#include <hip/hip_runtime.h>
#include <hip/hip_bf16.h>

typedef __attribute__((ext_vector_type(2))) float v2f;
typedef __attribute__((ext_vector_type(8))) float v8f;

#define FEAT      128
#define NBOX      512
#define SCALE     1.25f            /* 1/(0.1*8) */
#define PC_MIN_C  (-51.2f)
#define OVERLAP   0.1f
#define EPSF      1.1920928955078125e-7f  /* FLT_EPSILON */

// Block tile: 32 rows x 64 cols. 8 waves/block, each wave one 16x16 subtile (2x4 grid).
// Grid: 8 batches * (4 y-tiles * 2 x-tiles) = 64 blocks.
__global__ __launch_bounds__(256)
void bev_heatmap_kernel(const float* __restrict__ boxes,
                        const float* __restrict__ cvals,
                        float* __restrict__ out)
{
    __shared__ __align__(16) float  s_raw[NBOX * 7 + NBOX]; // 4096 dwords: boxes then cv
    __shared__ __align__(16) float4 s_list[NBOX];           // culled {meta, inv, k, pad}
    __shared__ int s_cnt;

    const int tid = threadIdx.x;
    const int b   = blockIdx.x >> 3;
    const int t   = blockIdx.x & 7;
    const int ty0 = (t >> 1) * 32;     // 4 tiles of 32 rows
    const int tx0 = (t & 1) * 64;      // 2 tiles of 64 cols

    // ---------------- Phase 1: async-stage raw batch data into LDS ----------------
    // 896 x 16B chunks of boxes + 128 x 16B chunks of center_values = 1024 chunks.
    const char* boxesB = (const char*)(boxes + (size_t)b * NBOX * 7);
    const char* cvB    = (const char*)(cvals + (size_t)b * NBOX);
#pragma unroll
    for (int j = 0; j < 4; ++j) {
        int c = tid + j * 256;
        unsigned lds_addr = (unsigned)(size_t)(&s_raw[c * 4]); // low 32b of flat = LDS byte off
        unsigned long long ga = (c < 896)
            ? (unsigned long long)(boxesB + (size_t)c * 16)
            : (unsigned long long)(cvB + (size_t)(c - 896) * 16);
        asm volatile("global_load_async_to_lds_b128 %0, %1, off"
                     :: "v"(lds_addr), "v"(ga) : "memory");
    }
    asm volatile("s_wait_asynccnt 0" ::: "memory");
    if (tid == 0) s_cnt = 0;
    __syncthreads();

    // ------- Phase 2: per-box params + tile cull, compact records (ds_add_rtn) -----
    for (int n = tid; n < NBOX; n += 256) {
        const float* bx = &s_raw[n * 7];
        float x = bx[0], y = bx[1];
        float w_fm = bx[3] * SCALE;
        float l_fm = bx[4] * SCALE;
        float h = l_fm, wd = w_fm;            // gaussian_radius(height=l_fm, width=w_fm)

        float b1 = h + wd;
        float c1 = wd * h * (1.f - OVERLAP) / (1.f + OVERLAP);
        float r1 = 0.5f * (b1 + sqrtf(fmaxf(b1 * b1 - 4.f * c1, 0.f)));

        float b2 = 2.f * (h + wd);
        float c2 = (1.f - OVERLAP) * wd * h;
        float r2 = 0.5f * (b2 + sqrtf(fmaxf(b2 * b2 - 16.f * c2, 0.f)));

        float b3 = -2.f * OVERLAP * (h + wd);
        float c3 = (OVERLAP - 1.f) * wd * h;
        float r3 = 0.5f * (b3 + sqrtf(fmaxf(b3 * b3 - 16.f * OVERLAP * c3, 0.f)));

        float rf = fminf(r1, fminf(r2, r3));
        int   r  = max(2, (int)floorf(rf));

        int cx = (int)((x - PC_MIN_C) * SCALE);   // trunc == astype(int32)
        int cy = (int)((y - PC_MIN_C) * SCALE);
        bool valid = (w_fm > 0.f) && (l_fm > 0.f) &&
                     (cx >= 0) && (cx < FEAT) && (cy >= 0) && (cy < FEAT);

        float sig = (2.f * (float)r + 1.f) * (1.f / 6.f);
        float inv = 1.f / (2.f * sig * sig);
        float k   = valid ? s_raw[NBOX * 7 + n] : 0.f;

        // cull against this block's tile (with radius margin); k>0 implies valid,
        // so cx,cy in [0,128) and r <= ~25 -> 8-bit fields are safe.
        if (k > 0.f &&
            cx + r >= tx0 && cx - r <= tx0 + 63 &&
            cy + r >= ty0 && cy - r <= ty0 + 31) {
            int p = atomicAdd(&s_cnt, 1);   // ds_add_rtn_u32
            float4 rec;
            rec.x = __int_as_float((cx & 0xFF) | ((cy & 0xFF) << 8) | ((r & 0xFF) << 16));
            rec.y = inv;
            rec.z = k;
            rec.w = 0.f;
            s_list[p] = rec;                // ds_store_b128
        }
    }
    __syncthreads();
    const int cnt = s_cnt;

    // ---------------- Phase 3: per-wave 16x16 subtile, WMMA outer-sum dist2 -------
    const int lane   = tid & 31;
    const int wave   = tid >> 5;
    const int lo16   = lane & 15;
    const int hi8    = (lane >> 4) << 3;         // 0 or 8
    const bool lol   = (lane < 16);
    const int oy     = ty0 + (wave >> 2) * 16;   // 2 subtile rows
    const int ox     = tx0 + (wave & 3) * 16;    // 4 subtile cols
    const int yA     = oy + lo16;                // row index feeding A-matrix
    const int xB     = ox + lo16;                // col index feeding B-matrix / output

    float acc[8];
#pragma unroll
    for (int rr = 0; rr < 8; ++rr) acc[rr] = 0.f;
    v8f czero = {0.f, 0.f, 0.f, 0.f, 0.f, 0.f, 0.f, 0.f};

    for (int i = 0; i < cnt; ++i) {
        float4 rec = s_list[i];                  // one uniform ds_load_b128
        int meta = __float_as_int(rec.x);
        int cx = meta & 0xFF;
        int cy = (meta >> 8) & 0xFF;
        int r  = (meta >> 16) & 0xFF;
        float inv = rec.y;
        float k   = rec.z;

        float dy = (float)(yA - cy);
        float dx = (float)(xB - cx);

        // A (16x4): col0 = dy^2, col1 = 1, cols 2,3 = 0
        v2f A;  A.x  = lol ? dy * dy : 0.f;  A.y  = lol ? 1.f : 0.f;
        // B (4x16): row0 = 1, row1 = dx^2, rows 2,3 = 0
        v2f Bm; Bm.x = lol ? 1.f : 0.f;      Bm.y = lol ? dx * dx : 0.f;

        // D[y][x] = dy^2 + dx^2 for the whole 16x16 tile in one XDL op
        v8f d = __builtin_amdgcn_wmma_f32_16x16x4_f32(
                    false, A, false, Bm, (short)0, czero, false, false);

        // fold the row-invariant column test into k once per box (branch-free)
        int adx = xB - cx; adx = adx < 0 ? -adx : adx;
        float kk = (adx <= r) ? k : 0.f;

#pragma unroll
        for (int rr = 0; rr < 8; ++rr) {
            int yy  = oy + hi8 + rr;               // C/D layout: M = rr + hi8
            int ady = yy - cy; ady = ady < 0 ? -ady : ady;
            float g = __expf(-d[rr] * inv);        // v_exp_f32 (unconditional)
            float v = g * kk;
            v = (g >= EPSF)  ? v : 0.f;            // independent selects, no exec
            v = (ady <= r)   ? v : 0.f;            // branching around the exp
            acc[rr] = fmaxf(acc[rr], v);
        }
    }

    // ---------------- Phase 4: write heat tile ------------------------------------
    float* outB = out + (size_t)b * FEAT * FEAT;
#pragma unroll
    for (int rr = 0; rr < 8; ++rr) {
        int yy = oy + hi8 + rr;
        outB[yy * FEAT + xB] = acc[rr];
    }
}

extern "C" void kernel_launch(void* const* d_in, const int* in_sizes, int n_in,
                              void* d_out, int out_size, void* d_ws, size_t ws_size,
                              hipStream_t stream) {
    (void)in_sizes; (void)n_in; (void)out_size; (void)d_ws; (void)ws_size;
    const float* boxes = (const float*)d_in[0];   // [8, 512, 7]
    const float* cv    = (const float*)d_in[1];   // [8, 512]
    float*       out   = (float*)d_out;           // [8, 1, 128, 128]
    bev_heatmap_kernel<<<64, 256, 0, stream>>>(boxes, cv, out);
}